// GAT_47107201303135
// MI455X (gfx1250) — compile-verified
//
#include <hip/hip_runtime.h>
#include <hip/hip_bf16.h>

typedef __attribute__((ext_vector_type(2))) float v2f;
typedef __attribute__((ext_vector_type(8))) float v8f;

#define GAT_NEG_SLOPE 0.2f

// ---------------------------------------------------------------- fill ------
__global__ void fill_u32_kernel(unsigned int* p, unsigned int v, long long n) {
  long long t = (long long)blockIdx.x * blockDim.x + threadIdx.x;
  if (t < n) p[t] = v;
}

// ------------------------------------------- fp32 GEMM via V_WMMA_F32_16X16X4
// C[M,Nc] = A[M,K] * B[K,Nc], all row-major fp32.
// Requires M%16==0, K%4==0, Nc%64==0 (true here: M=50000, K=256, Nc=256/128).
// One wave -> one 16x64 output tile (4 WMMA n-tiles, A-fragment reused 4x).
// Register double-buffered: fragment loads for step k+1 are issued before the
// step-k WMMAs so global loads overlap the chained matrix ops.
// A frag (32-bit 16x4, ISA 7.12.2): lanes 0-15 row M=lane hold K=k,k+1;
// lanes 16-31 row M=lane-16 hold K=k+2,k+3. B assumed symmetric (lane = col).
// C/D: VGPR i -> row m_base + i (+8 for lanes 16-31), col = lane&15.
__global__ void gemm_f32_wmma_kernel(const float* __restrict__ A,
                                     const float* __restrict__ B,
                                     float* __restrict__ C,
                                     int M, int K, int Nc) {
  const int NT = 4;
  int wave = (int)((blockIdx.x * (long long)blockDim.x + threadIdx.x) >> 5);
  int lane = threadIdx.x & 31;
  int n_groups = Nc / (16 * NT);
  int m_tiles = M / 16;
  if (wave >= m_tiles * n_groups) return;   // wave-uniform: EXEC stays all-1s
  int m_tile = wave / n_groups;
  int n_group = wave - m_tile * n_groups;

  int row  = m_tile * 16 + (lane & 15);
  int kh   = (lane >> 4) << 1;              // 0 (lanes 0-15) or 2 (lanes 16-31)
  int col0 = n_group * (16 * NT) + (lane & 15);

  const float* ap = A + (long long)row * K + kh;        // advances by 4
  const float* bp = B + (long long)kh * Nc + col0;      // advances by 4*Nc

  v8f acc[NT] = {};

  // preload k = 0 fragments
  v2f a_cur; a_cur.x = ap[0]; a_cur.y = ap[1];          // contiguous -> b64
  v2f b_cur[NT];
#pragma unroll
  for (int t = 0; t < NT; ++t) {
    b_cur[t].x = bp[t * 16];
    b_cur[t].y = bp[Nc + t * 16];
  }

  int steps = K >> 2;
  for (int it = 1; it < steps; ++it) {
    ap += 4;
    bp += (long long)Nc << 2;
    // issue next-step loads first; they overlap the WMMAs below
    v2f a_nxt; a_nxt.x = ap[0]; a_nxt.y = ap[1];
    v2f b_nxt[NT];
#pragma unroll
    for (int t = 0; t < NT; ++t) {
      b_nxt[t].x = bp[t * 16];
      b_nxt[t].y = bp[Nc + t * 16];
    }
#pragma unroll
    for (int t = 0; t < NT; ++t)
      acc[t] = __builtin_amdgcn_wmma_f32_16x16x4_f32(
          false, a_cur, false, b_cur[t], (short)0, acc[t], false, false);
    a_cur = a_nxt;
#pragma unroll
    for (int t = 0; t < NT; ++t) b_cur[t] = b_nxt[t];
  }
#pragma unroll
  for (int t = 0; t < NT; ++t)
    acc[t] = __builtin_amdgcn_wmma_f32_16x16x4_f32(
        false, a_cur, false, b_cur[t], (short)0, acc[t], false, false);

  int rbase = m_tile * 16 + ((lane >> 4) << 3);
#pragma unroll
  for (int t = 0; t < NT; ++t) {
    float* cp = C + (long long)rbase * Nc + col0 + t * 16;
#pragma unroll
    for (int i = 0; i < 8; ++i) cp[(long long)i * Nc] = acc[t][i];
  }
}

// ------------------------------------------------ per-node attention scores -
// a_s[n,h] = dot(h[n,h,:], att_src[h,:]);  a_d likewise. Wave per node.
__global__ void attn_scores_kernel(const float* __restrict__ h,
                                   const float* __restrict__ att_src,
                                   const float* __restrict__ att_dst,
                                   float* __restrict__ a_s,
                                   float* __restrict__ a_d,
                                   int Nn, int H, int D) {
  int wave = (int)((blockIdx.x * (long long)blockDim.x + threadIdx.x) >> 5);
  int lane = threadIdx.x & 31;
  if (wave >= Nn) return;
  for (int hh = 0; hh < H; ++hh) {
    float ss = 0.f, sd = 0.f;
    for (int dd = lane; dd < D; dd += 32) {
      float v = h[(long long)wave * H * D + hh * D + dd];
      ss += v * att_src[hh * D + dd];
      sd += v * att_dst[hh * D + dd];
    }
    for (int off = 16; off > 0; off >>= 1) {
      ss += __shfl_down(ss, off);
      sd += __shfl_down(sd, off);
    }
    if (lane == 0) {
      a_s[wave * H + hh] = ss;
      a_d[wave * H + hh] = sd;
    }
  }
}

// ------------------------------------------------------------- edge helpers -
__device__ __forceinline__ void edge_endpoints(const int* __restrict__ ei,
                                               int e, int E, int& s, int& d) {
  if (e < E) { s = ei[e]; d = ei[E + e]; }
  else       { s = d = e - E; }            // appended self-loops
}
// monotone float <-> signed-int order mapping (handles negatives / -inf)
__device__ __forceinline__ int ord_of_f32(float f) {
  int i = __float_as_int(f);
  return (i >= 0) ? i : (i ^ 0x7FFFFFFF);
}
__device__ __forceinline__ float f32_of_ord(int i) {
  return __int_as_float((i >= 0) ? i : (i ^ 0x7FFFFFFF));
}

// pass A: alpha = leaky_relu(a_s[src]+a_d[dst]); segment max via atomicMax(int)
__global__ void edge_alpha_max_kernel(const float* __restrict__ a_s,
                                      const float* __restrict__ a_d,
                                      const int* __restrict__ ei,
                                      int E, int Nn, int H,
                                      float* __restrict__ alpha,
                                      int* __restrict__ m_ord) {
  int t = blockIdx.x * blockDim.x + threadIdx.x;
  int Etot = E + Nn;
  if (t >= Etot * H) return;
  int e = t / H, hh = t - (t / H) * H;
  int s, d; edge_endpoints(ei, e, E, s, d);
  float al = a_s[s * H + hh] + a_d[d * H + hh];
  al = (al > 0.f) ? al : al * GAT_NEG_SLOPE;
  alpha[t] = al;
  atomicMax(&m_ord[d * H + hh], ord_of_f32(al));
}

// pass B: e = exp(alpha - m[dst]); segment sum (in-place over alpha buffer)
__global__ void edge_exp_sum_kernel(const int* __restrict__ ei,
                                    int E, int Nn, int H,
                                    const int* __restrict__ m_ord,
                                    float* __restrict__ alpha_e,
                                    float* __restrict__ denom) {
  int t = blockIdx.x * blockDim.x + threadIdx.x;
  int Etot = E + Nn;
  if (t >= Etot * H) return;
  int e = t / H, hh = t - (t / H) * H;
  int s, d; edge_endpoints(ei, e, E, s, d);
  (void)s;
  float ev = expf(alpha_e[t] - f32_of_ord(m_ord[d * H + hh]));
  alpha_e[t] = ev;
  atomicAdd(&denom[d * H + hh], ev);
}

// pass C: out[dst] += h[src] * e/(denom[dst]+1e-16). Wave per edge; lanes
// stream the feature row; atomics land in L2 (out fits in 192MB L2).
__global__ void edge_aggregate_kernel(const float* __restrict__ hfeat,
                                      const float* __restrict__ e_buf,
                                      const float* __restrict__ denom,
                                      const int* __restrict__ ei,
                                      int E, int Nn, int H, int D,
                                      float* __restrict__ out) {
  int wave = (int)((blockIdx.x * (long long)blockDim.x + threadIdx.x) >> 5);
  int lane = threadIdx.x & 31;
  int Etot = E + Nn;
  if (wave >= Etot) return;
  int s, d; edge_endpoints(ei, wave, E, s, d);
  int C = H * D;
  for (int hh = 0; hh < H; ++hh) {
    float coef = e_buf[wave * H + hh] / (denom[d * H + hh] + 1e-16f);
    const float* hp = hfeat + (long long)s * C + hh * D;
    float*       op = out   + (long long)d * C + hh * D;
    for (int c = lane; c < D; c += 32)
      atomicAdd(&op[c], hp[c] * coef);
  }
}

// epilogue: out += bias; optional ELU
__global__ void bias_act_kernel(float* __restrict__ out,
                                const float* __restrict__ bias,
                                long long total, int C, int do_elu) {
  long long t = (long long)blockIdx.x * blockDim.x + threadIdx.x;
  if (t >= total) return;
  float v = out[t] + bias[(int)(t % C)];
  if (do_elu) v = (v > 0.f) ? v : (expf(v) - 1.0f);
  out[t] = v;
}

// ---------------------------------------------------------------------------
extern "C" void kernel_launch(void* const* d_in, const int* in_sizes, int n_in,
                              void* d_out, int out_size, void* d_ws, size_t ws_size,
                              hipStream_t stream) {
  (void)n_in; (void)out_size; (void)ws_size;
  const float* x    = (const float*)d_in[0];
  const int*   ei   = (const int*)d_in[1];     // [2,E] flat: row0=src, row1=dst
  const float* W1   = (const float*)d_in[2];
  const float* as1w = (const float*)d_in[3];
  const float* ad1w = (const float*)d_in[4];
  const float* b1   = (const float*)d_in[5];
  const float* W2   = (const float*)d_in[6];
  const float* as2w = (const float*)d_in[7];
  const float* ad2w = (const float*)d_in[8];
  const float* b2   = (const float*)d_in[9];

  const int IN = 256, HID = 128, H = 2, OUT = 128;
  const int Nn   = in_sizes[0] / IN;
  const int E    = in_sizes[1] / 2;
  const int Etot = E + Nn;
  const int C1   = H * HID;                    // 256

  // workspace carve-up (floats)
  float* p    = (float*)d_ws;
  float* h1   = p; p += (long long)Nn * C1;
  float* agg1 = p; p += (long long)Nn * C1;    // becomes x2 after bias+ELU
  float* h2   = p; p += (long long)Nn * OUT;
  float* as1  = p; p += Nn * H;
  float* ad1  = p; p += Nn * H;
  float* den1 = p; p += Nn * H;
  int*   m1   = (int*)p; p += Nn * H;
  float* as2  = p; p += Nn;
  float* ad2  = p; p += Nn;
  float* den2 = p; p += Nn;
  int*   m2   = (int*)p; p += Nn;
  float* al1  = p; p += (long long)Etot * H;
  float* al2  = p; p += Etot;
  float* outp = (float*)d_out;

  const int BS = 256;
  auto cdiv = [](long long a, long long b) { return (int)((a + b - 1) / b); };
  const unsigned ORD_NEG_INF = 0x807FFFFFu;    // ord_of_f32(-inf)

  // ------------------------------- layer 1: GATConv(256 -> 128, heads=2) ----
  {
    long long waves = (long long)(Nn / 16) * (C1 / 64);
    gemm_f32_wmma_kernel<<<cdiv(waves * 32, BS), BS, 0, stream>>>(x, W1, h1, Nn, IN, C1);
  }
  attn_scores_kernel<<<cdiv((long long)Nn * 32, BS), BS, 0, stream>>>(h1, as1w, ad1w, as1, ad1, Nn, H, HID);
  fill_u32_kernel<<<cdiv((long long)Nn * H, BS), BS, 0, stream>>>((unsigned*)m1, ORD_NEG_INF, (long long)Nn * H);
  fill_u32_kernel<<<cdiv((long long)Nn * H, BS), BS, 0, stream>>>((unsigned*)den1, 0u, (long long)Nn * H);
  fill_u32_kernel<<<cdiv((long long)Nn * C1, BS), BS, 0, stream>>>((unsigned*)agg1, 0u, (long long)Nn * C1);
  edge_alpha_max_kernel<<<cdiv((long long)Etot * H, BS), BS, 0, stream>>>(as1, ad1, ei, E, Nn, H, al1, m1);
  edge_exp_sum_kernel<<<cdiv((long long)Etot * H, BS), BS, 0, stream>>>(ei, E, Nn, H, m1, al1, den1);
  edge_aggregate_kernel<<<cdiv((long long)Etot * 32, BS), BS, 0, stream>>>(h1, al1, den1, ei, E, Nn, H, HID, agg1);
  bias_act_kernel<<<cdiv((long long)Nn * C1, BS), BS, 0, stream>>>(agg1, b1, (long long)Nn * C1, C1, 1);

  // ------------------------------- layer 2: GATConv(256 -> 128, heads=1) ----
  {
    long long waves = (long long)(Nn / 16) * (OUT / 64);
    gemm_f32_wmma_kernel<<<cdiv(waves * 32, BS), BS, 0, stream>>>(agg1, W2, h2, Nn, C1, OUT);
  }
  attn_scores_kernel<<<cdiv((long long)Nn * 32, BS), BS, 0, stream>>>(h2, as2w, ad2w, as2, ad2, Nn, 1, OUT);
  fill_u32_kernel<<<cdiv((long long)Nn, BS), BS, 0, stream>>>((unsigned*)m2, ORD_NEG_INF, (long long)Nn);
  fill_u32_kernel<<<cdiv((long long)Nn, BS), BS, 0, stream>>>((unsigned*)den2, 0u, (long long)Nn);
  fill_u32_kernel<<<cdiv((long long)Nn * OUT, BS), BS, 0, stream>>>((unsigned*)outp, 0u, (long long)Nn * OUT);
  edge_alpha_max_kernel<<<cdiv((long long)Etot, BS), BS, 0, stream>>>(as2, ad2, ei, E, Nn, 1, al2, m2);
  edge_exp_sum_kernel<<<cdiv((long long)Etot, BS), BS, 0, stream>>>(ei, E, Nn, 1, m2, al2, den2);
  edge_aggregate_kernel<<<cdiv((long long)Etot * 32, BS), BS, 0, stream>>>(h2, al2, den2, ei, E, Nn, 1, OUT, outp);
  bias_act_kernel<<<cdiv((long long)Nn * OUT, BS), BS, 0, stream>>>(outp, b2, (long long)Nn * OUT, OUT, 0);
}